// SingleAttentionHead_89060441850566
// MI455X (gfx1250) — compile-verified
//
#include <hip/hip_runtime.h>
#include <math.h>

#define B_   256
#define N_   256
#define L_   1024
#define D_   256
#define KQ_  64
#define OUT_ 64
#define SSTR 1028   // padded LDS row stride in floats: m*1028 %64 = 4m -> conflict-free b64 reads

typedef float v2f __attribute__((ext_vector_type(2)));
typedef float v8f __attribute__((ext_vector_type(8)));

// D = A(16x4) * B(4x16) + C, fp32, wave32.
static __device__ __forceinline__ v8f wmma4(v2f a, v2f b, v8f c) {
  return __builtin_amdgcn_wmma_f32_16x16x4_f32(
      /*neg_a=*/false, a, /*neg_b=*/false, b,
      /*c_mod=*/(short)0, c, /*reuse_a=*/false, /*reuse_b=*/false);
}

// ---------------------------------------------------------------------------
// Kernel 1: q_s = (queries @ Wq) * 1/sqrt(KQ).  64 wave-tiles of 16x16.
// ---------------------------------------------------------------------------
__global__ void qproj_kernel(const float* __restrict__ q,
                             const float* __restrict__ Wq,
                             float* __restrict__ q_s) {
  const int wid  = threadIdx.x >> 5;
  const int lane = threadIdx.x & 31;
  const int w    = blockIdx.x * 8 + wid;   // 0..63
  const int rt   = w >> 2;                 // 16-row tile (0..15)
  const int ct   = w & 3;                  // 16-col tile (0..3)
  const int m    = lane & 15;
  const int half = lane >> 4;

  const float* arow = q + (size_t)(rt * 16 + m) * D_;
  v8f c = {};
  for (int k = 0; k < D_; k += 4) {
    v2f a, b;
    a.x = arow[k + 2 * half];
    a.y = arow[k + 2 * half + 1];
    b.x = Wq[(size_t)(k + 2 * half)     * KQ_ + ct * 16 + m];
    b.y = Wq[(size_t)(k + 2 * half + 1) * KQ_ + ct * 16 + m];
    c = wmma4(a, b, c);
  }
#pragma unroll
  for (int j = 0; j < 8; ++j)
    q_s[(size_t)(rt * 16 + j + 8 * half) * KQ_ + ct * 16 + m] = c[j] * 0.125f;
}

// ---------------------------------------------------------------------------
// Kernel 2: k_proj = keys @ Wk, v_proj = values @ Wv (grid.y selects which).
// One wave computes a 16x64 strip: A-loads reused across 4 output col tiles.
// ---------------------------------------------------------------------------
__global__ void kvproj_kernel(const float* __restrict__ keys,
                              const float* __restrict__ values,
                              const float* __restrict__ Wk,
                              const float* __restrict__ Wv,
                              float* __restrict__ kp,
                              float* __restrict__ vp) {
  const float* X; const float* W; float* O;
  if (blockIdx.y == 0) { X = keys;   W = Wk; O = kp; }
  else                 { X = values; W = Wv; O = vp; }

  const int wid  = threadIdx.x >> 5;
  const int lane = threadIdx.x & 31;
  const int m    = lane & 15;
  const int half = lane >> 4;
  const size_t strip = (size_t)blockIdx.x * 8 + wid;  // 0..16383 (16-row strips of N*L)

  const float* arow = X + (strip * 16 + m) * D_;
  v8f c0 = {}, c1 = {}, c2 = {}, c3 = {};
  for (int k = 0; k < D_; k += 4) {
    v2f a;
    a.x = arow[k + 2 * half];
    a.y = arow[k + 2 * half + 1];
    const float* w0 = W + (size_t)(k + 2 * half)     * KQ_;
    const float* w1 = W + (size_t)(k + 2 * half + 1) * KQ_;
    v2f b;
    b.x = w0[m];      b.y = w1[m];      c0 = wmma4(a, b, c0);
    b.x = w0[16 + m]; b.y = w1[16 + m]; c1 = wmma4(a, b, c1);
    b.x = w0[32 + m]; b.y = w1[32 + m]; c2 = wmma4(a, b, c2);
    b.x = w0[48 + m]; b.y = w1[48 + m]; c3 = wmma4(a, b, c3);
  }
  float* orow = O + strip * 16 * KQ_;
#pragma unroll
  for (int j = 0; j < 8; ++j) {
    const size_t r = (size_t)(j + 8 * half) * KQ_;
    orow[r +      m] = c0[j];
    orow[r + 16 + m] = c1[j];
    orow[r + 32 + m] = c2[j];
    orow[r + 48 + m] = c3[j];
  }
}

// ---------------------------------------------------------------------------
// Kernel 3 (fused): per (n, 32-row b-tile):
//   logits[32][1024] in LDS  ->  softmax (wave32 shfl reductions, write attn)
//   -> result tile = softmax @ v_proj[n], consumed straight from LDS.
// ---------------------------------------------------------------------------
__global__ void attn_kernel(const float* __restrict__ q_s,
                            const float* __restrict__ kp,
                            const float* __restrict__ vp,
                            float* __restrict__ out_res,
                            float* __restrict__ out_attn) {
  extern __shared__ float s[];   // [32][SSTR]
  const int n    = blockIdx.y;
  const int b0   = blockIdx.x * 32;
  const int wid  = threadIdx.x >> 5;
  const int lane = threadIdx.x & 31;
  const int m    = lane & 15;
  const int half = lane >> 4;

  const float* kpn = kp + (size_t)n * L_ * KQ_;
  const float* vpn = vp + (size_t)n * L_ * OUT_;

  // --- Stage A: s[b][l] = q_s[b0+b] . kpn[l]  (K = 64) ---
  const float* a0row = q_s + (size_t)(b0 + m) * KQ_;
  const float* a1row = a0row + 16 * KQ_;
  for (int lt = wid; lt < L_ / 16; lt += 8) {   // uniform 8 iters/wave, no divergence
    const int l0 = lt * 16;
    const float* brow = kpn + (size_t)(l0 + m) * KQ_;  // B[k][l] = kp[l][k]
    v8f c0 = {}, c1 = {};
    for (int k = 0; k < KQ_; k += 4) {
      v2f b, a;
      b.x = brow[k + 2 * half];  b.y = brow[k + 2 * half + 1];
      a.x = a0row[k + 2 * half]; a.y = a0row[k + 2 * half + 1];
      c0 = wmma4(a, b, c0);
      a.x = a1row[k + 2 * half]; a.y = a1row[k + 2 * half + 1];
      c1 = wmma4(a, b, c1);
    }
#pragma unroll
    for (int j = 0; j < 8; ++j) {
      s[(size_t)(j + 8 * half) * SSTR + l0 + m]        = c0[j];
      s[(size_t)(16 + j + 8 * half) * SSTR + l0 + m]   = c1[j];
    }
  }
  __syncthreads();

  // --- Stage B: softmax over L per row; write attn[n, b0+r, :] ---
  for (int r = wid * 4; r < wid * 4 + 4; ++r) {
    float* row = s + (size_t)r * SSTR;
    float mx = -1e30f;
    for (int l = lane; l < L_; l += 32) mx = fmaxf(mx, row[l]);
#pragma unroll
    for (int off = 16; off > 0; off >>= 1) mx = fmaxf(mx, __shfl_xor(mx, off, 32));
    float sum = 0.f;
    for (int l = lane; l < L_; l += 32) {
      float e = __expf(row[l] - mx);
      row[l] = e;
      sum += e;
    }
#pragma unroll
    for (int off = 16; off > 0; off >>= 1) sum += __shfl_xor(sum, off, 32);
    const float inv = 1.f / sum;
    float* aout = out_attn + ((size_t)n * B_ + (b0 + r)) * L_;
    for (int l = lane; l < L_; l += 32) {
      const float v = row[l] * inv;
      row[l] = v;
      aout[l] = v;
    }
  }
  __syncthreads();

  // --- Stage C: result[b0..b0+31, n, :] = s(32x1024) @ vpn(1024x64) ---
  const int sub_b = wid >> 2;          // 0..1
  const int o0    = (wid & 3) * 16;    // output col tile
  const float* srow = s + (size_t)(sub_b * 16 + m) * SSTR;
  v8f c = {};
  for (int k = 0; k < L_; k += 4) {
    v2f a, b;
    a.x = srow[k + 2 * half];
    a.y = srow[k + 2 * half + 1];
    b.x = vpn[(size_t)(k + 2 * half)     * OUT_ + o0 + m];
    b.y = vpn[(size_t)(k + 2 * half + 1) * OUT_ + o0 + m];
    c = wmma4(a, b, c);
  }
#pragma unroll
  for (int j = 0; j < 8; ++j) {
    const int r = sub_b * 16 + j + 8 * half;
    out_res[(size_t)(b0 + r) * (N_ * OUT_) + (size_t)n * OUT_ + o0 + m] = c[j];
  }
}

// ---------------------------------------------------------------------------
extern "C" void kernel_launch(void* const* d_in, const int* in_sizes, int n_in,
                              void* d_out, int out_size, void* d_ws, size_t ws_size,
                              hipStream_t stream) {
  const float* queries = (const float*)d_in[0];
  const float* keys    = (const float*)d_in[1];
  const float* values  = (const float*)d_in[2];
  const float* Wq      = (const float*)d_in[3];
  const float* Wk      = (const float*)d_in[4];
  const float* Wv      = (const float*)d_in[5];

  float* out_res  = (float*)d_out;                       // [B, N, OUT]
  float* out_attn = out_res + (size_t)B_ * N_ * OUT_;    // [N, B, L]

  float* q_s = (float*)d_ws;                             // [B, KQ]
  float* kp  = q_s + (size_t)B_ * KQ_;                   // [N, L, KQ]
  float* vp  = kp + (size_t)N_ * L_ * KQ_;               // [N, L, OUT]

  const int smem = 32 * SSTR * (int)sizeof(float);       // 131,584 B < 320 KB WGP LDS
  (void)hipFuncSetAttribute(reinterpret_cast<const void*>(attn_kernel),
                            hipFuncAttributeMaxDynamicSharedMemorySize, smem);

  hipLaunchKernelGGL(qproj_kernel, dim3(8), dim3(256), 0, stream, queries, Wq, q_s);
  hipLaunchKernelGGL(kvproj_kernel, dim3(2048, 2), dim3(256), 0, stream,
                     keys, values, Wk, Wv, kp, vp);
  hipLaunchKernelGGL(attn_kernel, dim3(B_ / 32, N_), dim3(256), smem, stream,
                     q_s, kp, vp, out_res, out_attn);
}